// SelfContactSmall_80221399154694
// MI455X (gfx1250) — compile-verified
//
#include <hip/hip_runtime.h>
#include <cmath>

// ---------------------------------------------------------------------------
// SelfContact min-distance on MI455X (gfx1250, wave32).
//
// v2v[b,i,j] = ||x_i||^2 + ||x_j||^2 - 2 x_i.x_j   (geomask false -> +inf)
// out0[b,j]  = min_i v2v[b,i,j]
// out1[b,j]  = out0 < 0.02^2  (as 0.0/1.0 float)
//
// One V_WMMA_F32_16X16X4_F32 computes a full 16x16 squared-distance tile:
//   A[i,:] = (-2x_i, -2y_i, -2z_i, sq_i)   (16x4 f32, reloaded per tile)
//   B[:,j] = ( x_j,   y_j,   z_j,  1  )    (4x16 f32, loop-invariant)
//   C[i,j] = sq_j                          (broadcast, loop-invariant)
//   D      = A*B + C  ==  sq_i + sq_j - 2 x_i.x_j
//
// Bandwidth analysis: geomask dominates (N^2 ~ 110 MB bool, read once per
// batch ~ 220 MB -> ~10 us at 23.3 TB/s; it nearly fits the 192 MB L2).
// To actually reach that bound we need enough waves in flight to hide load
// latency: 8 waves per 16-column strip split the row loop (10,480 waves
// total ~ 10/SIMD), then combine partial column minima through LDS.
// ---------------------------------------------------------------------------

typedef __attribute__((ext_vector_type(2))) float v2f;
typedef __attribute__((ext_vector_type(8))) float v8f;

#define EUCL_THR2 0.0004f   // 0.02^2
#define INFF __builtin_inff()
#define WAVES_PER_BLOCK 8

// Precompute per-vertex squared norms once (each sq[i] is reused by ~655
// column-strip blocks).
__global__ void sq_norm_kernel(const float* __restrict__ verts,
                               float* __restrict__ sq, int total) {
  int idx = blockIdx.x * blockDim.x + threadIdx.x;
  if (idx < total) {
    const float x = verts[idx * 3 + 0];
    const float y = verts[idx * 3 + 1];
    const float z = verts[idx * 3 + 2];
    sq[idx] = fmaf(x, x, fmaf(y, y, z * z));
  }
}

// 256 threads = 8 wave32s per block. Block owns one 16-column strip of one
// batch; wave w handles row tiles t = w, w+8, ... EXEC is all-ones around
// every WMMA (predication via clamped indices / selects only).
__global__ __launch_bounds__(256) void contact_min_kernel(
    const float* __restrict__ verts,          // [B, N, 3] f32
    const unsigned char* __restrict__ mask,   // [N, N] bool (1 byte, numpy)
    const float* __restrict__ sq,             // [B, N] f32 (workspace)
    float* __restrict__ out,                  // [2, B, N] f32
    int N, int B) {
  const int b    = blockIdx.y;
  const int j0   = blockIdx.x << 4;           // first column of this strip
  const int tid  = threadIdx.x;
  const int wave = tid >> 5;                  // 0..7
  const int lane = tid & 31;                  // wave32 lane
  const int col  = lane & 15;                 // output column within strip
  const int hi   = lane >> 4;                 // 0: K0/K1 & rows 0-7; 1: K2/K3 & rows 8-15

  __shared__ float red[WAVES_PER_BLOCK][16];  // per-wave partial column minima

  const float* __restrict__ vb  = verts + (size_t)b * N * 3;
  const float* __restrict__ sqb = sq + (size_t)b * N;

  // ---- loop-invariant B tile: B[:,j] = (x_j, y_j, z_j, 1) -----------------
  // B 4x16 f32 layout mirrors A 16x4: VGPR0 = {lanes0-15: K=0 row,
  // lanes16-31: K=2 row}; VGPR1 = {lanes0-15: K=1, lanes16-31: K=3}.
  int jc = j0 + col;
  if (jc >= N) jc = N - 1;                    // clamp; OOB cols never stored
  const float xj = vb[jc * 3 + 0];
  const float yj = vb[jc * 3 + 1];
  const float zj = vb[jc * 3 + 2];
  v2f bvec;
  bvec.x = hi ? zj : xj;
  bvec.y = hi ? 1.0f : yj;

  // ---- loop-invariant C tile: C[i,j] = sq_j (same for every row) ----------
  const float sqj = sqb[jc];
  v8f cacc;
#pragma unroll
  for (int r = 0; r < 8; ++r) cacc[r] = sqj;

  // running column minima: acc[r] covers rows {16*t + 8*hi + r}
  v8f acc;
#pragma unroll
  for (int r = 0; r < 8; ++r) acc[r] = INFF;

  const int numRowTiles = (N + 15) >> 4;
  const int rowBase = hi << 3;                // 0 or 8

  for (int t = wave; t < numRowTiles; t += WAVES_PER_BLOCK) {
    const int i0 = t << 4;

    // ---- A tile: A[i,:] = (-2x, -2y, -2z, sq_i); OOB rows forced to +inf --
    // lane<16 holds (K0,K1)=(-2x,-2y); lane>=16 holds (K2,K3)=(-2z, sq_i).
    int ic = i0 + col;
    const bool oob = (ic >= N);
    if (oob) ic = N - 1;                      // clamped load address
    const float f0 = vb[ic * 3 + (hi ? 2 : 0)];        // x or z
    const float f1 = hi ? sqb[ic] : vb[ic * 3 + 1];    // sq_i or y
    v2f a;
    a.x = oob ? 0.0f : (-2.0f * f0);
    a.y = oob ? (hi ? INFF : 0.0f) : (hi ? f1 : (-2.0f * f1));
    // OOB row: (0,0,0,inf).(x,y,z,1) + sq_j = inf  (no 0*inf -> no NaN)

    // ---- full 16x16 squared-distance tile in one instruction --------------
    v8f d = __builtin_amdgcn_wmma_f32_16x16x4_f32(
        /*neg_a=*/false, a, /*neg_b=*/false, bvec,
        /*c_mod=*/(short)0, cacc, /*reuse_a=*/false, /*reuse_b=*/false);

    // ---- geomask + min-accumulate; D[r] is row (i0 + rowBase + r) ---------
#pragma unroll
    for (int r = 0; r < 8; ++r) {
      int ir = i0 + rowBase + r;
      if (ir >= N) ir = N - 1;                // clamp load; value is inf anyway
      const unsigned char m = mask[(size_t)ir * (size_t)N + (size_t)jc];
      const float dv = m ? d[r] : INFF;       // v_cndmask
      acc[r] = fminf(acc[r], dv);
    }

    // prefetch this wave's next tile's mask rows (t+8 -> +128 rows)
    const int pr = i0 + (WAVES_PER_BLOCK << 4) + col;
    if (pr < N) {
      __builtin_prefetch(mask + (size_t)pr * (size_t)N + (size_t)j0, 0, 3);
    }
  }

  // ---- per-wave fold: 8 rows per lane, then the two row halves ------------
  float mn = acc[0];
#pragma unroll
  for (int r = 1; r < 8; ++r) mn = fminf(mn, acc[r]);
  const float other = __shfl_xor(mn, 16, 32); // lane l <-> l+16
  mn = fminf(mn, other);

  if (hi == 0) red[wave][col] = mn;           // lanes 0-15 of each wave
  __syncthreads();

  // ---- cross-wave fold + store (first 16 threads of the block) ------------
  if (tid < 16) {
    float m = red[0][tid];
#pragma unroll
    for (int w = 1; w < WAVES_PER_BLOCK; ++w) m = fminf(m, red[w][tid]);
    const int j = j0 + tid;
    if (j < N) {
      out[(size_t)b * N + j] = m;                                // v2v_min
      out[(size_t)B * N + (size_t)b * N + j] =
          (m < EUCL_THR2) ? 1.0f : 0.0f;                         // in_contact
    }
  }
}

extern "C" void kernel_launch(void* const* d_in, const int* in_sizes, int n_in,
                              void* d_out, int out_size, void* d_ws,
                              size_t ws_size, hipStream_t stream) {
  const float* verts        = (const float*)d_in[0];         // [B, N, 3] f32
  const unsigned char* mask = (const unsigned char*)d_in[1]; // [N, N] bool

  // Recover shapes: in_sizes[1] = N*N, in_sizes[0] = B*N*3.
  const int N = (int)(std::sqrt((double)in_sizes[1]) + 0.5);
  const int totalVerts = in_sizes[0] / 3;  // B*N
  const int B = totalVerts / N;

  float* sq = (float*)d_ws;  // B*N floats of scratch

  sq_norm_kernel<<<(totalVerts + 255) / 256, 256, 0, stream>>>(verts, sq,
                                                               totalVerts);

  dim3 grid((N + 15) / 16, B);
  contact_min_kernel<<<grid, 256, 0, stream>>>(verts, mask, sq, (float*)d_out,
                                               N, B);
}